// g_sgcn_54709293416497
// MI455X (gfx1250) — compile-verified
//
#include <hip/hip_runtime.h>
#include <hip/hip_bf16.h>

// SDE Euler-Maruyama scan on MI455X (gfx1250), fp32 WMMA 16x16x4 path, v2.
//
// Roofline: noise stream = 424 MB (one pass, ~18us @ 23.3 TB/s) dominates ->
// loaded non-temporally. State (852 KB x2 layouts), adj_pad (173 KB), weights
// are L2-resident (192 MB L2) and re-hit every one of the 1000 sequential
// steps. All WMMA operands are laid out so each lane's A/B fragment is one
// unpredicated 8-byte load: A needs (row fixed, K contiguous) -> natural
// row-major; B needs (col fixed, K contiguous) -> transposed copies
// (yT / WfT / WgT / WoutT). Zero-padding N 207->208 removes all bounds checks
// from the K-loops.

typedef float v2f __attribute__((ext_vector_type(2)));
typedef float v8f __attribute__((ext_vector_type(8)));

#define B_DIM   8
#define N_DIM   207
#define NP      208          // padded N (multiple of 4/8)
#define H_DIM   64
#define N_TILES 13           // ceil(207/16)
#define OUT_DIM 1024         // 32*32

__device__ __forceinline__ v8f wmma_f32_k4(v2f a, v2f b, v8f c) {
    // D = A(16x4,f32) x B(4x16,f32) + C(16x16,f32)
    return __builtin_amdgcn_wmma_f32_16x16x4_f32(
        false, a, false, b, (short)0, c, false, false);
}

__device__ __forceinline__ v2f ldg2(const float* p) { return *(const v2f*)p; }

// ---------------- prep kernels (rebuild padded/transposed ws each call) ----

__global__ void g_prep_pad_adj(const float* __restrict__ adj,
                               float* __restrict__ adj_pad) {
    int i = blockIdx.x * blockDim.x + threadIdx.x;   // over NP*NP
    if (i < NP * NP) {
        int r = i / NP, c = i % NP;
        adj_pad[i] = (r < N_DIM && c < N_DIM) ? adj[r * N_DIM + c] : 0.0f;
    }
}

__global__ void g_prep_transpose(const float* __restrict__ W,
                                 float* __restrict__ WT, int K, int C) {
    int i = blockIdx.x * blockDim.x + threadIdx.x;   // W[K][C] -> WT[C][K]
    if (i < K * C) {
        int k = i / C, c = i % C;
        WT[c * K + k] = W[i];
    }
}

__global__ void g_prep_state(const float* __restrict__ x,
                             float* __restrict__ y,    // [B][NP][H]
                             float* __restrict__ yT) { // [B][H][NP]
    int i = blockIdx.x * blockDim.x + threadIdx.x;    // over B*NP*H
    if (i < B_DIM * NP * H_DIM) {
        int b = i / (NP * H_DIM);
        int r = (i / H_DIM) % NP;
        int h = i % H_DIM;
        float v = (r < N_DIM) ? x[(b * N_DIM + r) * H_DIM + h] : 0.0f;
        y[i] = v;
        yT[(b * H_DIM + h) * NP + r] = v;
    }
}

// ---------------- fused Euler-Maruyama step ---------------------------------
// Block = 128 = 4 waves; wave w owns the 16(n) x 16(h) tile at h0 = 16w.
__global__ __launch_bounds__(128) void g_sde_step_kernel(
    const float* __restrict__ y_in,     // [B][NP][H]
    const float* __restrict__ yT_in,    // [B][H][NP]
    float* __restrict__ y_out,
    float* __restrict__ yT_out,
    const float* __restrict__ adj_pad,  // [NP][NP]
    const float* __restrict__ WfT,      // [H][H]  (WfT[c][k] = Wf[k][c])
    const float* __restrict__ bf,
    const float* __restrict__ WgT,
    const float* __restrict__ bg,
    const float* __restrict__ noise_t)  // [B][N_DIM][H] (unpadded!)
{
    __shared__ float agg_s[16][66];     // stride 66: even + conflict-free

    const int lane  = threadIdx.x & 31;
    const int wave  = threadIdx.x >> 5;
    const int b     = blockIdx.x / N_TILES;
    const int n0    = (blockIdx.x % N_TILES) * 16;
    const int h0    = wave * 16;

    const int row16 = lane & 15;        // M (A) / N (B,C,D) index in tile
    const int k2    = (lane >> 4) * 2;  // lane half -> K base {0,2}

    const float* yb  = y_in  + (size_t)b * (NP * H_DIM);
    const float* yTb = yT_in + (size_t)b * (H_DIM * NP);

    // ---- 1) agg = adj @ y : K over m, 208 wide, two accumulators
    {
        const float* ap = adj_pad + (n0 + row16) * NP + k2;     // A row
        const float* bp = yTb + (h0 + row16) * NP + k2;         // B col (yT row)
        v8f acc0 = {}, acc1 = {};
        for (int m0 = 0; m0 < NP; m0 += 8) {
            v2f a0 = ldg2(ap + m0),     b0 = ldg2(bp + m0);
            v2f a1 = ldg2(ap + m0 + 4), b1 = ldg2(bp + m0 + 4);
            acc0 = wmma_f32_k4(a0, b0, acc0);
            acc1 = wmma_f32_k4(a1, b1, acc1);
        }
        v8f acc = acc0 + acc1;
        // spill 16x16 agg tile to LDS (C layout: row = v + (lane/16)*8)
#pragma unroll
        for (int v = 0; v < 8; ++v)
            agg_s[v + (k2 >> 1) * 8][h0 + row16] = acc[v];
    }
    __syncthreads();

    // ---- 2) f-pre = agg @ Wf : K = 64
    v8f fa;
    {
        const float* bp = WfT + (h0 + row16) * H_DIM + k2;
        v8f acc0 = {}, acc1 = {};
#pragma unroll
        for (int k0 = 0; k0 < H_DIM; k0 += 8) {
            v2f a0 = *(const v2f*)&agg_s[row16][k0 + k2];
            v2f a1 = *(const v2f*)&agg_s[row16][k0 + 4 + k2];
            v2f b0 = ldg2(bp + k0), b1 = ldg2(bp + k0 + 4);
            acc0 = wmma_f32_k4(a0, b0, acc0);
            acc1 = wmma_f32_k4(a1, b1, acc1);
        }
        fa = acc0 + acc1;
    }

    // ---- 3) g-pre = y @ Wg : K = 64 (row 207 of y is padding -> masked later)
    v8f ga;
    {
        const float* ap = yb + (n0 + row16) * H_DIM + k2;
        const float* bp = WgT + (h0 + row16) * H_DIM + k2;
        v8f acc0 = {}, acc1 = {};
#pragma unroll
        for (int k0 = 0; k0 < H_DIM; k0 += 8) {
            v2f a0 = ldg2(ap + k0), b0 = ldg2(bp + k0);
            v2f a1 = ldg2(ap + k0 + 4), b1 = ldg2(bp + k0 + 4);
            acc0 = wmma_f32_k4(a0, b0, acc0);
            acc1 = wmma_f32_k4(a1, b1, acc1);
        }
        ga = acc0 + acc1;
    }

    // ---- 4) pointwise Euler-Maruyama update; keep pad row 207 at zero
    const float biasf   = bf[h0 + row16];
    const float biasg   = bg[h0 + row16];
    const float dt      = 0.001f;
    const float sqrt_dt = 0.03162277660168379f;
    float* yob        = y_out  + (size_t)b * (NP * H_DIM);
    float* yTob       = yT_out + (size_t)b * (H_DIM * NP);
    const float* nzb  = noise_t + (size_t)b * (N_DIM * H_DIM);
    const int   col   = h0 + row16;
#pragma unroll
    for (int v = 0; v < 8; ++v) {
        const int n = n0 + v + (k2 >> 1) * 8;
        float val = 0.0f;
        if (n < N_DIM) {
            const float f = tanhf(fa[v] + biasf);
            const float g = 1.0f / (1.0f + __expf(-(ga[v] + biasg)));
            const float z = __builtin_nontemporal_load(&nzb[n * H_DIM + col]);
            val = yb[n * H_DIM + col] + f * dt + g * (z * sqrt_dt);
        }
        yob[n * H_DIM + col] = val;       // padded: n==207 store is in-bounds
        yTob[col * NP + n]   = val;       // transposed copy for next step's B
    }
}

// ---------------- output: tanh(z1 @ W_out + b_out) --------------------------
// grid = (16, N_TILES, B); block = 128 (4 waves); wave w -> col tile bx*4+w.
__global__ __launch_bounds__(128) void g_sde_out_kernel(
    const float* __restrict__ z,        // [B][NP][H] padded final state
    const float* __restrict__ WoutT,    // [OUT_DIM][H]
    const float* __restrict__ b_out,    // [OUT_DIM]
    float* __restrict__ out)            // [B][N_DIM][OUT_DIM]
{
    const int lane  = threadIdx.x & 31;
    const int wave  = threadIdx.x >> 5;
    const int b     = blockIdx.z;
    const int n0    = blockIdx.y * 16;
    const int c0    = (blockIdx.x * 4 + wave) * 16;
    const int row16 = lane & 15;
    const int k2    = (lane >> 4) * 2;

    const float* ap = z + (size_t)b * (NP * H_DIM) + (n0 + row16) * H_DIM + k2;
    const float* bp = WoutT + (c0 + row16) * H_DIM + k2;

    v8f acc0 = {}, acc1 = {};
#pragma unroll
    for (int k0 = 0; k0 < H_DIM; k0 += 8) {
        v2f a0 = ldg2(ap + k0), b0 = ldg2(bp + k0);
        v2f a1 = ldg2(ap + k0 + 4), b1 = ldg2(bp + k0 + 4);
        acc0 = wmma_f32_k4(a0, b0, acc0);
        acc1 = wmma_f32_k4(a1, b1, acc1);
    }
    v8f acc = acc0 + acc1;

    const float bias = b_out[c0 + row16];
    float* ob = out + (size_t)b * (N_DIM * OUT_DIM);
#pragma unroll
    for (int v = 0; v < 8; ++v) {
        const int n = n0 + v + (k2 >> 1) * 8;
        if (n < N_DIM)
            ob[n * OUT_DIM + c0 + row16] = tanhf(acc[v] + bias);
    }
}

// ---------------- host-side launch ------------------------------------------
extern "C" void kernel_launch(void* const* d_in, const int* in_sizes, int n_in,
                              void* d_out, int out_size, void* d_ws, size_t ws_size,
                              hipStream_t stream) {
    (void)in_sizes; (void)n_in; (void)out_size; (void)ws_size;

    const float* x     = (const float*)d_in[0];  // [8,207,64]
    const float* adj   = (const float*)d_in[1];  // [207,207]
    const float* Wf    = (const float*)d_in[2];  // [64,64]
    const float* bf    = (const float*)d_in[3];  // [64]
    const float* Wg    = (const float*)d_in[4];  // [64,64]
    const float* bg    = (const float*)d_in[5];  // [64]
    const float* W_out = (const float*)d_in[6];  // [64,1024]
    const float* b_out = (const float*)d_in[7];  // [1024]
    const float* noise = (const float*)d_in[8];  // [1000,8,207,64]

    // workspace layout (floats)
    float* w = (float*)d_ws;
    const size_t state  = (size_t)B_DIM * NP * H_DIM;    // 106,496
    float* adj_pad = w;                 w += (size_t)NP * NP;        // 43,264
    float* WfT     = w;                 w += H_DIM * H_DIM;          //  4,096
    float* WgT     = w;                 w += H_DIM * H_DIM;          //  4,096
    float* WoutT   = w;                 w += (size_t)H_DIM * OUT_DIM;// 65,536
    float* y0      = w;                 w += state;
    float* y1      = w;                 w += state;
    float* yT0     = w;                 w += state;
    float* yT1     = w;                 w += state;

    const int TPB = 256;
    g_prep_pad_adj<<<(NP * NP + TPB - 1) / TPB, TPB, 0, stream>>>(adj, adj_pad);
    g_prep_transpose<<<(H_DIM * H_DIM + TPB - 1) / TPB, TPB, 0, stream>>>(
        Wf, WfT, H_DIM, H_DIM);
    g_prep_transpose<<<(H_DIM * H_DIM + TPB - 1) / TPB, TPB, 0, stream>>>(
        Wg, WgT, H_DIM, H_DIM);
    g_prep_transpose<<<(H_DIM * OUT_DIM + TPB - 1) / TPB, TPB, 0, stream>>>(
        W_out, WoutT, H_DIM, OUT_DIM);
    g_prep_state<<<((int)state + TPB - 1) / TPB, TPB, 0, stream>>>(x, y0, yT0);

    const dim3 blk(128);
    const dim3 grid_step(B_DIM * N_TILES);     // 104 blocks, 4 waves each
    const size_t nstate = (size_t)B_DIM * N_DIM * H_DIM;  // noise step stride

    const float *yi = y0, *yTi = yT0;
    float *yo = y1, *yTo = yT1;
    for (int t = 0; t < 1000; ++t) {
        g_sde_step_kernel<<<grid_step, blk, 0, stream>>>(
            yi, yTi, yo, yTo, adj_pad, WfT, bf, WgT, bg,
            noise + (size_t)t * nstate);
        const float* ti = yi; yi = yo; yo = (float*)ti;
        ti = yTi; yTi = yTo; yTo = (float*)ti;
    }

    const dim3 grid_out(16, N_TILES, B_DIM);   // 64 col tiles / 4 per block
    g_sde_out_kernel<<<grid_out, blk, 0, stream>>>(yi, WoutT, b_out, (float*)d_out);
}